// GNN_node_23270132810370
// MI455X (gfx1250) — compile-verified
//
#include <hip/hip_runtime.h>

#define N_NODES 50000
#define N_EDGES 800000
#define EMB_D   128
#define NUM_LAYER 5
#define NFEAT   9
#define MTILES  (N_NODES / 16)     // 3125, exact
#define PSTR    288                // paired-row LDS stride in floats: 288 mod 64 == 32
                                   // -> half-wave b64 fragment reads hit disjoint bank sets

typedef __attribute__((ext_vector_type(2))) float    v2f;
typedef __attribute__((ext_vector_type(8))) float    v8f;
typedef __attribute__((ext_vector_type(4))) int      v4i;
typedef __attribute__((ext_vector_type(8))) int      v8i;
typedef __attribute__((ext_vector_type(4))) unsigned v4u;

#if defined(__HIP_DEVICE_COMPILE__) && __has_builtin(__builtin_amdgcn_tensor_load_to_lds) && __has_builtin(__builtin_amdgcn_s_wait_tensorcnt)
#define USE_TDM 1
#else
#define USE_TDM 0
#endif

// ---------------- atom encode: h[n][d] = sum_f atom_emb[f][x[n][f]][d] ----------------
__global__ void encode_kernel(const int* __restrict__ x,
                              const float* __restrict__ atom_emb,
                              float* __restrict__ h) {
    int t = blockIdx.x * blockDim.x + threadIdx.x;
    int node = t >> 5;
    if (node >= N_NODES) return;
    int d = (t & 31) * 4;
    float4 acc = make_float4(0.f, 0.f, 0.f, 0.f);
    const int* xr = x + node * NFEAT;
#pragma unroll
    for (int f = 0; f < NFEAT; ++f) {
        int idx = xr[f];
        float4 e = *(const float4*)(atom_emb + ((size_t)(f * 128 + idx)) * EMB_D + d);
        acc.x += e.x; acc.y += e.y; acc.z += e.z; acc.w += e.w;
    }
    *(float4*)(h + (size_t)node * EMB_D + d) = acc;
}

// ---------------- zero agg ----------------
__global__ void zero_kernel(float4* __restrict__ p, int n4) {
    int t = blockIdx.x * blockDim.x + threadIdx.x;
    if (t < n4) p[t] = make_float4(0.f, 0.f, 0.f, 0.f);
}

// ---------------- edge scatter: agg[dst] += h[src]  (one wave per edge) ----------------
__global__ void scatter_kernel(const int* __restrict__ ei,
                               const float* __restrict__ h,
                               float* __restrict__ agg) {
    int t = blockIdx.x * blockDim.x + threadIdx.x;
    int e = t >> 5;
    if (e >= N_EDGES) return;
    int d = (t & 31) * 4;
    int s  = ei[e];
    int dn = ei[N_EDGES + e];
    float4 v = *(const float4*)(h + (size_t)s * EMB_D + d);
    float* ap = agg + (size_t)dn * EMB_D + d;
    unsafeAtomicAdd(ap + 0, v.x);   // -> global_atomic_add_f32 (L2-resident)
    unsafeAtomicAdd(ap + 1, v.y);
    unsafeAtomicAdd(ap + 2, v.z);
    unsafeAtomicAdd(ap + 3, v.w);
}

// ---------------- pack weights into K-paired layout ----------------
// Wp[(k>>1)*256 + n*2 + (k&1)] = W[k*128 + n]; both K-components of a B-fragment
// become one aligned 8-byte unit -> single ds_load_b64 into an even VGPR pair.
__global__ void pack_w_kernel(const float* __restrict__ Wrel,
                              const float* __restrict__ Wroot,
                              float* __restrict__ Wp) {
    int t = blockIdx.x * blockDim.x + threadIdx.x;
    if (t >= 2 * EMB_D * EMB_D) return;
    const float* src = (t < EMB_D * EMB_D) ? Wrel : Wroot;
    float*       dst = Wp + ((t < EMB_D * EMB_D) ? 0 : EMB_D * EMB_D);
    int i = t & (EMB_D * EMB_D - 1);
    int k = i >> 7, n = i & 127;
    dst[(k >> 1) * 256 + n * 2 + (k & 1)] = src[i];
}

#if USE_TDM
// 1-D TDM DMA of one 4096-dword paired-weight chunk into LDS, with hardware
// padding (interval=256 dwords, amount=32 dwords) producing 288-float LDS rows.
__device__ __forceinline__ void tdm_load_chunk(const float* gsrc, unsigned ldsByteOff) {
    unsigned long long ga = (unsigned long long)(uintptr_t)gsrc;
    v4u g0;
    g0[0] = 1u;                                           // count = 1 (valid descriptor)
    g0[1] = ldsByteOff;                                   // lds_addr
    g0[2] = (unsigned)(ga & 0xffffffffu);                 // global_addr[31:0]
    g0[3] = (unsigned)((ga >> 32) & 0x1ffffffu) | (2u << 30); // ga[56:32] | type=2
    v8i g1 = (v8i)0;
    g1[0] = (int)((2u << 16)        // data_size = 4B
                | (1u << 20)        // pad_enable
                | (7u << 22)        // pad_interval: 256 dwords
                | (31u << 25));     // pad_amount:   32 dwords
    g1[1] = (int)(4096u << 16);     // tensor_dim0 = 4096 (bits 79:48, low half)
    g1[2] = (int)(1u << 16);        // tensor_dim1 = 1
    g1[3] = (int)(4096u << 16);     // tile_dim0  = 4096 (bits 127:112)
    g1[4] = 1;                      // tile_dim1  = 1, tile_dim2 = 0
    g1[5] = 4096;                   // tensor_dim0_stride
    v4i z4 = (v4i)0;
    v8i z8 = (v8i)0;
    __builtin_amdgcn_tensor_load_to_lds(g0, g1, z4, z4, z8, 0);
}
#endif

// ---------------- fused dense update: hout = agg@Wrel + hin@Wroot + b ----------------
// 256 threads = 8 waves; wave owns one 16-row M-tile, all 128 output cols (8 N-tiles).
// Paired weights staged per 32-K chunk (TDM when available), V_WMMA_F32_16X16X4_F32 math.
// Per k-step: batch-load all 16 B fragments (one ds clause + one wait), then an
// unbroken burst of 16 WMMAs — removes the per-pair s_wait_dscnt bubbles.
__global__ __launch_bounds__(256) void gemm_kernel(const float* __restrict__ agg,
                                                   const float* __restrict__ hin,
                                                   const float* __restrict__ Wp,  // [2][16384] paired
                                                   const float* __restrict__ bias,
                                                   float* __restrict__ hout) {
    __shared__ __align__(16) float lWrel[16 * PSTR];   // 16 paired K-rows per chunk
    __shared__ __align__(16) float lWroot[16 * PSTR];

    const int tid   = threadIdx.x;
    const int wave  = tid >> 5;
    const int lane  = tid & 31;
    const int mtile = blockIdx.x * 8 + wave;
    const bool active = (mtile < MTILES);
    const int mbase = mtile * 16;
    const int mrow  = lane & 15;          // M (for A) / N (for B) lane index
    const int koff  = (lane >> 4) * 2;    // A frag: lanes 0-15 K=k0,k0+1; 16-31 K=k0+2,k0+3
    const int khalf = lane >> 4;          // B frag paired-row select

    v8f c[8];
#pragma unroll
    for (int i = 0; i < 8; ++i) c[i] = (v8f)0.0f;

    for (int kc = 0; kc < 4; ++kc) {      // K chunks of 32 (= 16 paired rows)
        __syncthreads();
#if USE_TDM
        if (wave == 0) {
            tdm_load_chunk(Wp + kc * 4096,
                           (unsigned)(uintptr_t)(&lWrel[0]));
            tdm_load_chunk(Wp + EMB_D * EMB_D + kc * 4096,
                           (unsigned)(uintptr_t)(&lWroot[0]));
            __builtin_amdgcn_s_wait_tensorcnt(0);
        }
#else
        {   // manual staging with the same 288-float padded rows
            const float* srel  = Wp + kc * 4096;
            const float* sroot = Wp + EMB_D * EMB_D + kc * 4096;
#pragma unroll
            for (int j = 0; j < 4; ++j) {
                int idx = (tid + j * 256) * 4;         // dword index 0..4095, 16B aligned
                int row = idx >> 8, col = idx & 255;
                *(float4*)(lWrel  + row * PSTR + col) = *(const float4*)(srel  + idx);
                *(float4*)(lWroot + row * PSTR + col) = *(const float4*)(sroot + idx);
            }
        }
#endif
        __syncthreads();

        if (active) {
#pragma unroll
            for (int ks = 0; ks < 8; ++ks) {          // 8 k-steps of 4 per chunk
                const int k0  = kc * 32 + ks * 4;
                const int k2l = ks * 2 + khalf;       // paired row within chunk
                v2f a_rel  = *(const v2f*)(agg + (size_t)(mbase + mrow) * EMB_D + k0 + koff);
                v2f a_root = *(const v2f*)(hin + (size_t)(mbase + mrow) * EMB_D + k0 + koff);
                const float* rrel  = lWrel  + k2l * PSTR;
                const float* rroot = lWroot + k2l * PSTR;
                // batch all 16 B fragments first -> one ds clause, one wait
                v2f brel[8], broot[8];
#pragma unroll
                for (int nt = 0; nt < 8; ++nt) {
                    const int n2 = (nt * 16 + mrow) * 2;
                    brel[nt]  = *(const v2f*)(rrel  + n2);   // ds_load_b64 (pairable 2addr)
                    broot[nt] = *(const v2f*)(rroot + n2);
                }
#pragma unroll
                for (int nt = 0; nt < 8; ++nt) {
                    c[nt] = __builtin_amdgcn_wmma_f32_16x16x4_f32(
                        false, a_rel, false, brel[nt], (short)0, c[nt], false, false);
                    c[nt] = __builtin_amdgcn_wmma_f32_16x16x4_f32(
                        false, a_root, false, broot[nt], (short)0, c[nt], false, false);
                }
            }
        }
    }

    if (active) {
        const int rowoff = (lane >> 4) * 8;   // C layout: lanes 16-31 hold M = v+8
#pragma unroll
        for (int nt = 0; nt < 8; ++nt) {
            float bv = bias[nt * 16 + mrow];
#pragma unroll
            for (int v = 0; v < 8; ++v) {
                hout[(size_t)(mbase + v + rowoff) * EMB_D + nt * 16 + mrow] = c[nt][v] + bv;
            }
        }
    }
}

extern "C" void kernel_launch(void* const* d_in, const int* in_sizes, int n_in,
                              void* d_out, int out_size, void* d_ws, size_t ws_size,
                              hipStream_t stream) {
    const int*   x        = (const int*)d_in[0];
    const int*   ei       = (const int*)d_in[1];
    // d_in[2] = edge_attr : unused by the reference
    const float* atom_emb = (const float*)d_in[3];
    const float* Wrel     = (const float*)d_in[4];
    const float* brel     = (const float*)d_in[5];
    const float* Wroot    = (const float*)d_in[6];
    float* out  = (float*)d_out;
    float* hbuf = (float*)d_ws;                        // N*D f32
    float* agg  = hbuf + (size_t)N_NODES * EMB_D;      // N*D f32
    float* Wp   = agg  + (size_t)N_NODES * EMB_D;      // 2*128*128 f32 paired weights

    const dim3 b256(256);
    encode_kernel<<<(N_NODES * 32 + 255) / 256, b256, 0, stream>>>(x, atom_emb, hbuf);

    for (int layer = 0; layer < NUM_LAYER; ++layer) {
        const float* hin = (layer % 2 == 0) ? hbuf : out;
        float*      hout = (layer % 2 == 0) ? out  : hbuf;   // layer 4 -> d_out
        zero_kernel<<<(N_NODES * EMB_D / 4 + 255) / 256, b256, 0, stream>>>(
            (float4*)agg, N_NODES * EMB_D / 4);
        scatter_kernel<<<(N_EDGES * 32 + 255) / 256, b256, 0, stream>>>(ei, hin, agg);
        pack_w_kernel<<<(2 * EMB_D * EMB_D + 255) / 256, b256, 0, stream>>>(
            Wrel + (size_t)layer * EMB_D * EMB_D,
            Wroot + (size_t)layer * EMB_D * EMB_D, Wp);
        gemm_kernel<<<(MTILES + 7) / 8, b256, 0, stream>>>(
            agg, hin, Wp, brel + (size_t)layer * EMB_D, hout);
    }
}